// RnCLoss_858993459320
// MI455X (gfx1250) — compile-verified
//
#include <hip/hip_runtime.h>
#include <hip/hip_bf16.h>

// RnC loss for N=512, D=128 fp32 on gfx1250.
// Stage 1: row squared-norms.
// Stage 2: Gram matrix via V_WMMA_F32_16X16X4_F32 (exact fp32 — avoids
//          cancellation error in x2_i + x2_j - 2*G), fused into sim = -d2/T.
// Stage 3: per-anchor masked logsumexp (O(N^2) per anchor, LDS broadcast).
// Stage 4: scalar reduce.

#define Nn 512
#define Dd 128
#define NEGV (-10000.0f)

typedef float v2f __attribute__((ext_vector_type(2)));
typedef float v8f __attribute__((ext_vector_type(8)));

__global__ void rnc_rownorm_kernel(const float* __restrict__ F,
                                   float* __restrict__ x2) {
    const int row = blockIdx.x * blockDim.x + threadIdx.x;
    if (row >= Nn) return;
    const float4* p = reinterpret_cast<const float4*>(F + row * Dd);
    float s = 0.f;
#pragma unroll
    for (int q = 0; q < Dd / 4; ++q) {
        float4 v = p[q];
        s = fmaf(v.x, v.x, s);
        s = fmaf(v.y, v.y, s);
        s = fmaf(v.z, v.z, s);
        s = fmaf(v.w, v.w, s);
    }
    x2[row] = s;
}

// One wave -> 16x32 output strip. 512 waves total (32 tilesM x 16 tilesN32).
__global__ void rnc_sim_wmma_kernel(const float* __restrict__ F,
                                    const float* __restrict__ x2,
                                    float* __restrict__ sim) {
    const int wave = (blockIdx.x * blockDim.x + threadIdx.x) >> 5;  // 0..511
    const int lane = threadIdx.x & 31;
    const int tm = wave >> 4;   // row tile, 16 rows
    const int tn = wave & 15;   // col strip, 32 cols
    const int r  = lane & 15;
    const int kb = (lane >> 4) << 1;  // 0 or 2 (ISA 16x4 f32 A/B layout)

    const float* arow  = F + (tm * 16 + r) * Dd + kb;
    const float* brow0 = F + (tn * 32 + r) * Dd + kb;
    const float* brow1 = brow0 + 16 * Dd;

    v8f c0 = {};
    v8f c1 = {};
#pragma unroll 8
    for (int k = 0; k < Dd; k += 4) {
        v2f a  = *reinterpret_cast<const v2f*>(arow + k);
        v2f b0 = *reinterpret_cast<const v2f*>(brow0 + k);
        v2f b1 = *reinterpret_cast<const v2f*>(brow1 + k);
        c0 = __builtin_amdgcn_wmma_f32_16x16x4_f32(
            false, a, false, b0, (short)0, c0, false, false);
        c1 = __builtin_amdgcn_wmma_f32_16x16x4_f32(
            false, a, false, b1, (short)0, c1, false, false);
    }

    // C/D layout: VGPR rr -> M=rr (lanes 0-15), M=rr+8 (lanes 16-31); N=lane&15
    const int rowOff = (lane >> 4) << 3;  // 0 or 8
    const int col0 = tn * 32 + (lane & 15);
    const float x2c0 = x2[col0];
    const float x2c1 = x2[col0 + 16];
#pragma unroll
    for (int rr = 0; rr < 8; ++rr) {
        const int row = tm * 16 + rr + rowOff;
        const float xr = x2[row];
        const float d0 = fmaxf(xr + x2c0 - 2.0f * c0[rr], 0.0f);
        const float d1 = fmaxf(xr + x2c1 - 2.0f * c1[rr], 0.0f);
        sim[row * Nn + col0]      = -0.5f * d0;  // T = 2 -> *0.5
        sim[row * Nn + col0 + 16] = -0.5f * d1;
    }
}

// One block per anchor i. Each thread handles j and j+256 with fused
// two-threshold max-pass + exp-pass over the LDS-resident row.
__global__ void rnc_anchor_loss_kernel(const float* __restrict__ sim,
                                       const float* __restrict__ labels,
                                       float* __restrict__ partial) {
    __shared__ float s_sim[Nn];
    __shared__ float s_ld[Nn];
    __shared__ float s_red[256];

    const int i = blockIdx.x;
    const int t = threadIdx.x;
    const float li = labels[i];
    const float* srow = sim + (size_t)i * Nn;

    for (int k = t; k < Nn; k += 256) {
        s_sim[k] = srow[k];
        s_ld[k]  = fabsf(labels[k] - li);
    }
    __syncthreads();

    const int j0 = t;
    const int j1 = t + 256;
    const float t0 = s_ld[j0];
    const float t1 = s_ld[j1];

    float m0 = NEGV, m1 = NEGV;
#pragma unroll 4
    for (int k = 0; k < Nn; ++k) {
        const float l  = s_ld[k];
        const float vb = (k == i) ? NEGV : s_sim[k];  // k != i exclusion
        m0 = fmaxf(m0, (l >= t0) ? vb : NEGV);
        m1 = fmaxf(m1, (l >= t1) ? vb : NEGV);
    }

    float s0 = 0.f, s1 = 0.f;
#pragma unroll 4
    for (int k = 0; k < Nn; ++k) {
        const float l  = s_ld[k];
        const float vb = (k == i) ? NEGV : s_sim[k];
        s0 += __expf(((l >= t0) ? vb : NEGV) - m0);
        s1 += __expf(((l >= t1) ? vb : NEGV) - m1);
    }

    const float loss0 = (j0 != i) ? (m0 + __logf(s0) - s_sim[j0]) : 0.f;
    const float loss1 = (j1 != i) ? (m1 + __logf(s1) - s_sim[j1]) : 0.f;

    s_red[t] = loss0 + loss1;
    __syncthreads();
#pragma unroll
    for (int off = 128; off > 0; off >>= 1) {
        if (t < off) s_red[t] += s_red[t + off];
        __syncthreads();
    }
    if (t == 0) partial[i] = s_red[0];
}

__global__ void rnc_finalize_kernel(const float* __restrict__ partial,
                                    float* __restrict__ out) {
    __shared__ float red[256];
    const int t = threadIdx.x;
    red[t] = partial[t] + partial[t + 256];
    __syncthreads();
#pragma unroll
    for (int off = 128; off > 0; off >>= 1) {
        if (t < off) red[t] += red[t + off];
        __syncthreads();
    }
    if (t == 0) out[0] = red[0] / (float)(Nn * (Nn - 1));
}

extern "C" void kernel_launch(void* const* d_in, const int* in_sizes, int n_in,
                              void* d_out, int out_size, void* d_ws, size_t ws_size,
                              hipStream_t stream) {
    const float* F      = (const float*)d_in[0];  // [512,128] fp32
    const float* labels = (const float*)d_in[1];  // [512]     fp32

    float* ws      = (float*)d_ws;
    float* sim     = ws;                 // N*N floats (1 MB)
    float* x2      = ws + Nn * Nn;       // N floats
    float* partial = x2 + Nn;            // N floats

    rnc_rownorm_kernel<<<Nn / 256, 256, 0, stream>>>(F, x2);
    rnc_sim_wmma_kernel<<<64, 256, 0, stream>>>(F, x2, sim);   // 512 waves
    rnc_anchor_loss_kernel<<<Nn, 256, 0, stream>>>(sim, labels, partial);
    rnc_finalize_kernel<<<1, 256, 0, stream>>>(partial, (float*)d_out);
}